// GCNAttention_1752346657005
// MI455X (gfx1250) — compile-verified
//
#include <hip/hip_runtime.h>
#include <hip/hip_bf16.h>

#define BIMG 256
#define CCAT 100
#define CPAD 112
#define NBOX 4096
#define PER  16
#define SLOTS 16
#define FEAT 2048
#define DIN  2052
#define DOUT 512

typedef __attribute__((ext_vector_type(2))) float v2f;
typedef __attribute__((ext_vector_type(8))) float v8f;

// ---------------------------------------------------------------------------
// K0: per-image classification of bboxes into (slot, occurrence) and build
// gather lists: row (b,slot) <- up to 3 (bbox, lin_w[occ]) entries.
// ---------------------------------------------------------------------------
__global__ void k0_classify(const int* __restrict__ label,
                            const float* __restrict__ linw,
                            int* __restrict__ row_bbox,
                            float* __restrict__ row_w,
                            int* __restrict__ cat_of_slot,
                            int* __restrict__ n_active) {
    __shared__ int lbl[PER], slt[PER], occ[PER];
    const int b = blockIdx.x;
    const int t = threadIdx.x;
    if (t < PER) lbl[t] = label[b * PER + t];
    __syncthreads();
    if (t < PER) {
        int o = 0, first = t;
        for (int j = t - 1; j >= 0; --j)
            if (lbl[j] == lbl[t]) { ++o; first = j; }
        int s = 0;
        for (int j = 0; j < first; ++j) {
            bool isfirst = true;
            for (int jj = 0; jj < j; ++jj)
                if (lbl[jj] == lbl[j]) { isfirst = false; break; }
            if (isfirst) ++s;
        }
        occ[t] = o; slt[t] = s;
    }
    __syncthreads();
    if (t < PER) {  // thread t owns slot-row t of this image
        for (int o = 0; o < 3; ++o) {
            int bb = -1;
            for (int j = 0; j < PER; ++j)
                if (slt[j] == t && occ[j] == o) bb = j;
            row_bbox[(b * SLOTS + t) * 3 + o] = (bb >= 0) ? (b * PER + bb) : -1;
            row_w[(b * SLOTS + t) * 3 + o] = linw[o];
        }
    }
    if (t == 0) {
        int na = 0;
        for (int j = 0; j < PER; ++j)
            if (occ[j] == 0) { cat_of_slot[b * SLOTS + slt[j]] = lbl[j]; ++na; }
        n_active[b] = na;
    }
}

// ---------------------------------------------------------------------------
// K1: delta_gi[row, :] = sum_o w_o * full[bbox_o, :]   (row = b*SLOTS + slot)
// full = [img_features | bbox_list].  No atomics: gather from 3-entry list.
// ---------------------------------------------------------------------------
__global__ void k1_delta_gi(const float* __restrict__ img,
                            const float* __restrict__ bbox,
                            const int* __restrict__ row_bbox,
                            const float* __restrict__ row_w,
                            float* __restrict__ dgi) {
    const int r = blockIdx.x;
    const int t = threadIdx.x;
    const int bb0 = row_bbox[r * 3 + 0];
    const int bb1 = row_bbox[r * 3 + 1];
    const int bb2 = row_bbox[r * 3 + 2];
    const float w0 = row_w[r * 3 + 0];
    const float w1 = row_w[r * 3 + 1];
    const float w2 = row_w[r * 3 + 2];
    for (int d = t; d < DIN; d += 256) {
        float acc = 0.f;
        if (bb0 >= 0) acc += w0 * (d < FEAT ? img[bb0 * FEAT + d] : bbox[bb0 * 4 + (d - FEAT)]);
        if (bb1 >= 0) acc += w1 * (d < FEAT ? img[bb1 * FEAT + d] : bbox[bb1 * 4 + (d - FEAT)]);
        if (bb2 >= 0) acc += w2 * (d < FEAT ? img[bb2 * FEAT + d] : bbox[bb2 * 4 + (d - FEAT)]);
        dgi[r * DIN + d] = acc;
    }
}

// ---------------------------------------------------------------------------
// K2: delta_support[4096,512] = delta_gi[4096,2052] @ gc_w[2052,512]
// fp32 WMMA 16x16x4. One wave per 16x16 output tile; 4 waves/block share the
// same M-tile (A rows hit in L0/L2 across waves), gc_w (4.2 MB) stays in L2.
// ISA layouts: A 16x4 -> lanes0-15 rows, VGPR0/1 = K {0,1} (lanes0-15) and
// K {2,3} (lanes16-31); B 4x16 mirrored; C/D: lane = column, VGPR r = row r/r+8.
// ---------------------------------------------------------------------------
__global__ void k2_gemm_dsup(const float* __restrict__ A,
                             const float* __restrict__ Bm,
                             float* __restrict__ Cm) {
    const int lane = threadIdx.x & 31;
    const int wave = threadIdx.x >> 5;
    const int ntile = blockIdx.x * 4 + wave;
    const int mtile = blockIdx.y;
    const int half = lane >> 4;        // 0 -> K{k,k+1}, 1 -> K{k+2,k+3}
    const int l16  = lane & 15;
    const float* __restrict__ arow = A + (size_t)(mtile * 16 + l16) * DIN + 2 * half;
    const float* __restrict__ bcol = Bm + ntile * 16 + l16 + (size_t)(2 * half) * DOUT;
    v8f acc = {};
    for (int k = 0; k < DIN; k += 4) {
        v2f a, b;
        a.x = arow[k];
        a.y = arow[k + 1];
        b.x = bcol[(size_t)k * DOUT];
        b.y = bcol[(size_t)(k + 1) * DOUT];
        acc = __builtin_amdgcn_wmma_f32_16x16x4_f32(false, a, false, b,
                                                    (short)0, acc, false, false);
    }
    const int row0 = mtile * 16 + half * 8;
    const int col  = ntile * 16 + l16;
#pragma unroll
    for (int r = 0; r < 8; ++r)
        Cm[(size_t)(row0 + r) * DOUT + col] = acc[r];
}

// ---------------------------------------------------------------------------
// K3: global_att[256, 100] = gfeat[256,2048] @ fc_w[100,2048]^T + fc_b
// fp32 WMMA, N padded to 112 with guarded B loads / stores.
// ---------------------------------------------------------------------------
__global__ void k3_gatt(const float* __restrict__ gfeat,
                        const float* __restrict__ fcw,
                        const float* __restrict__ fcb,
                        float* __restrict__ ga) {
    const int lane = threadIdx.x & 31;
    const int wave = threadIdx.x >> 5;
    const int mtile = blockIdx.y * 4 + wave;
    const int ntile = blockIdx.x;
    const int half = lane >> 4;
    const int l16  = lane & 15;
    const int col  = ntile * 16 + l16;
    const bool cok = (col < CCAT);
    const float* __restrict__ arow = gfeat + (size_t)(mtile * 16 + l16) * FEAT + 2 * half;
    const float* __restrict__ brow = fcw + (size_t)(cok ? col : 0) * FEAT + 2 * half;
    v8f acc = {};
    for (int k = 0; k < FEAT; k += 4) {
        v2f a, b;
        a.x = arow[k];
        a.y = arow[k + 1];
        b.x = cok ? brow[k] : 0.f;       // B[k,n] = fc_w[n*FEAT + k]
        b.y = cok ? brow[k + 1] : 0.f;
        acc = __builtin_amdgcn_wmma_f32_16x16x4_f32(false, a, false, b,
                                                    (short)0, acc, false, false);
    }
    if (cok) {
        const float bias = fcb[col];
        const int row0 = mtile * 16 + half * 8;
#pragma unroll
        for (int r = 0; r < 8; ++r)
            ga[(size_t)(row0 + r) * CPAD + col] = acc[r] + bias;
    }
}

// ---------------------------------------------------------------------------
// K4: const_support[e] = lin_b * sum_d gc_w[d,e] + gc_b[e]
// ---------------------------------------------------------------------------
__global__ void k4_csup(const float* __restrict__ gcw,
                        const float* __restrict__ gcb,
                        const float* __restrict__ linb,
                        float* __restrict__ cs) {
    const int e = blockIdx.x * 256 + threadIdx.x;
    if (e >= DOUT) return;
    float s = 0.f;
    for (int d = 0; d < DIN; ++d) s += gcw[(size_t)d * DOUT + e];
    cs[e] = linb[0] * s + gcb[e];
}

// ---------------------------------------------------------------------------
// K5: fused  out[b,e] = sum_i ga[b,i] * lrelu( rowsum(adj')[b,i]*cs[e]
//                          + sum_s adj'[b,i,cat_s] * dsup[b,s,e] )
// adj' = X + adj.  All per-image operands staged in ~42 KB LDS.
// ---------------------------------------------------------------------------
__global__ void k5_out(const float* __restrict__ adj,
                       const float* __restrict__ X,
                       const float* __restrict__ ga,
                       const float* __restrict__ cs,
                       const float* __restrict__ dsup,
                       const int* __restrict__ cat_of_slot,
                       const int* __restrict__ n_active,
                       float* __restrict__ out) {
    __shared__ float s_cs[DOUT];
    __shared__ float s_ds[SLOTS * DOUT];
    __shared__ float s_ga[CCAT];
    __shared__ float s_rs[CCAT];
    __shared__ float s_w[CCAT * SLOTS];
    const int b = blockIdx.x;
    const int t = threadIdx.x;
    const int na = n_active[b];

    s_cs[t] = cs[t];
    s_cs[t + 256] = cs[t + 256];
    for (int i = t; i < SLOTS * DOUT; i += 256)
        s_ds[i] = dsup[(size_t)b * SLOTS * DOUT + i];
    if (t < CCAT) {
        s_ga[t] = ga[b * CPAD + t];
        const float* __restrict__ arow = adj + (size_t)(b * CCAT + t) * CCAT;
        const float* __restrict__ xrow = X + t * CCAT;
        float s = 0.f;
        for (int j = 0; j < CCAT; ++j) s += xrow[j] + arow[j];
        s_rs[t] = s;
    }
    for (int idx = t; idx < CCAT * SLOTS; idx += 256) {
        const int i = idx / SLOTS, s = idx % SLOTS;
        float w = 0.f;
        if (s < na) {
            const int cat = cat_of_slot[b * SLOTS + s];
            w = X[i * CCAT + cat] + adj[(size_t)(b * CCAT + i) * CCAT + cat];
        }
        s_w[idx] = w;
    }
    __syncthreads();

    const int e0 = t, e1 = t + 256;
    float acc0 = 0.f, acc1 = 0.f;
    for (int i = 0; i < CCAT; ++i) {
        const float rs = s_rs[i];
        float v0 = rs * s_cs[e0];
        float v1 = rs * s_cs[e1];
        const float* __restrict__ wrow = &s_w[i * SLOTS];
        for (int s = 0; s < na; ++s) {
            const float w = wrow[s];
            v0 += w * s_ds[s * DOUT + e0];
            v1 += w * s_ds[s * DOUT + e1];
        }
        v0 = v0 > 0.f ? v0 : 0.01f * v0;
        v1 = v1 > 0.f ? v1 : 0.01f * v1;
        const float g = s_ga[i];
        acc0 += g * v0;
        acc1 += g * v1;
    }
    out[(size_t)b * DOUT + e0] = acc0;
    out[(size_t)b * DOUT + e1] = acc1;
}

// ---------------------------------------------------------------------------
extern "C" void kernel_launch(void* const* d_in, const int* in_sizes, int n_in,
                              void* d_out, int out_size, void* d_ws, size_t ws_size,
                              hipStream_t stream) {
    const float* img   = (const float*)d_in[0];
    const float* bbox  = (const float*)d_in[1];
    const int*   label = (const int*)  d_in[2];
    // d_in[3] = batch (implied by layout; unused)
    const float* gfeat = (const float*)d_in[4];
    const float* adj   = (const float*)d_in[5];
    const float* X     = (const float*)d_in[6];
    const float* linw  = (const float*)d_in[7];
    const float* linb  = (const float*)d_in[8];
    const float* fcw   = (const float*)d_in[9];
    const float* fcb   = (const float*)d_in[10];
    const float* gcw   = (const float*)d_in[11];
    const float* gcb   = (const float*)d_in[12];
    float* out = (float*)d_out;

    // workspace layout (floats, then ints)
    float* ws = (float*)d_ws;
    const size_t n_rows = (size_t)BIMG * SLOTS;              // 4096
    float* dgi  = ws;                                        // 4096*2052
    float* dsup = dgi + n_rows * DIN;                        // 4096*512
    float* ga   = dsup + n_rows * DOUT;                      // 256*112
    float* cs   = ga + (size_t)BIMG * CPAD;                  // 512
    float* roww = cs + DOUT;                                 // 4096*3
    int* ibase  = (int*)(roww + n_rows * 3);
    int* rowbb  = ibase;                                     // 4096*3
    int* catsl  = rowbb + n_rows * 3;                        // 4096
    int* nact   = catsl + n_rows;                            // 256

    k0_classify<<<BIMG, 32, 0, stream>>>(label, linw, rowbb, roww, catsl, nact);
    k1_delta_gi<<<(int)n_rows, 256, 0, stream>>>(img, bbox, rowbb, roww, dgi);
    k2_gemm_dsup<<<dim3(DOUT / 16 / 4, n_rows / 16), 128, 0, stream>>>(dgi, gcw, dsup);
    k3_gatt<<<dim3(CPAD / 16, BIMG / 16 / 4), 128, 0, stream>>>(gfeat, fcw, fcb, ga);
    k4_csup<<<(DOUT + 255) / 256, 256, 0, stream>>>(gcw, gcb, linb, cs);
    k5_out<<<BIMG, 256, 0, stream>>>(adj, X, ga, cs, dsup, catsl, nact, out);
}